// YoloLoss_15668040696394
// MI455X (gfx1250) — compile-verified
//
#include <hip/hip_runtime.h>
#include <math.h>

#define NB_    16
#define NA_    5
#define NH_    96
#define NW_    96
#define MAXT_  50
#define NC_    40
#define CH_    45                      // 5 + NUM_CLASSES
#define NCELL  (NB_*NA_*NH_*NW_)       // 737280
#define THREADS 256
#define NBLK2  (NCELL/THREADS)         // 2880 (exact)
#define NQ     10
#define SCALE_ 16.0f

__device__ __constant__ float c_AW[NA_] = {1.0f, 2.0f, 4.0f, 2.0f, 4.0f}; // ANCHORS_PX[:,0]/SCALE
__device__ __constant__ float c_AH[NA_] = {1.0f, 2.0f, 4.0f, 4.0f, 2.0f}; // ANCHORS_PX[:,1]/SCALE

typedef __attribute__((address_space(1))) int gint_as;   // global int*  ("__device__ int*")
typedef __attribute__((address_space(3))) int lint_as;   // LDS int*     ("__shared__ int*")

// ---------------------------------------------------------------- init flags
// flag byte: bit0 = mask (init 0), bit1 = conf_mask (init 1) -> 0x02
__global__ void init_flags_kernel(unsigned int* __restrict__ flags_u32) {
    int i = blockIdx.x * blockDim.x + threadIdx.x;   // NCELL/4 words
    if (i < NCELL / 4) flags_u32[i] = 0x02020202u;
}

// ---------------------------------------------------------------- build targets
// One thread per batch replays the reference scan sequentially (last-write-wins).
__global__ void build_targets_kernel(const float* __restrict__ pred,
                                     const float* __restrict__ target,
                                     const int*   __restrict__ tsizes,
                                     unsigned char* __restrict__ flags,
                                     unsigned char* __restrict__ tlab,
                                     float* __restrict__ tx, float* __restrict__ ty,
                                     float* __restrict__ tw, float* __restrict__ th,
                                     int* __restrict__ bGT, int* __restrict__ bCor) {
    int b = blockIdx.x;
    if (b >= NB_) return;
    int ts = tsizes[b];
    if (ts > MAXT_) ts = MAXT_;
    if (ts < 0) ts = 0;
    int nCor = 0;

    for (int t = 0; t < ts; ++t) {
        const float* row = target + ((size_t)b * MAXT_ + t) * (13 + NC_);
        float gx = row[0] / SCALE_;
        float gy = row[1] / SCALE_;
        float gh = row[3] / SCALE_;
        float gw = row[4] / SCALE_;
        int gi = (int)gx;            // positive -> trunc == astype(int32)
        int gj = (int)gy;

        // anchor IoUs + argmax (first max wins)
        float ious[NA_];
        float best = -1e30f; int bn = 0;
        float gArea = (gw + 1.0f) * (gh + 1.0f);
        for (int a = 0; a < NA_; ++a) {
            float iw = fmaxf(fminf(gw, c_AW[a]) + 1.0f, 0.0f);
            float ih = fmaxf(fminf(gh, c_AH[a]) + 1.0f, 0.0f);
            float inter = iw * ih;
            float aArea = (c_AW[a] + 1.0f) * (c_AH[a] + 1.0f);
            float iou = inter / (gArea + aArea - inter + 1e-16f);
            ious[a] = iou;
            if (iou > best) { best = iou; bn = a; }
        }

        int ci = gi < 0 ? 0 : (gi > NW_ - 1 ? NW_ - 1 : gi);
        int cj = gj < 0 ? 0 : (gj > NH_ - 1 ? NH_ - 1 : gj);
        bool inb = (ci == gi) && (cj == gj);

        if (inb) {
            // conf_mask zeroing where iou > IGNORE_THRES
            for (int a = 0; a < NA_; ++a) {
                if (ious[a] > 0.5f) {
                    size_t c = (((size_t)b * NA_ + a) * NH_ + gj) * NW_ + gi;
                    flags[c] = (unsigned char)(flags[c] & ~2);
                }
            }
            size_t c = (((size_t)b * NA_ + bn) * NH_ + gj) * NW_ + gi;
            flags[c] = (unsigned char)(flags[c] | 3);   // mask=1, conf_mask=1
            tx[c] = gx - (float)gi;
            ty[c] = gy - (float)gj;
            tw[c] = logf(gw / c_AW[bn] + 1e-16f);
            th[c] = logf(gh / c_AH[bn] + 1e-16f);
            // label = argmax of one-hot
            int lbl = 0; float lm = row[13];
            for (int k = 1; k < NC_; ++k)
                if (row[13 + k] > lm) { lm = row[13 + k]; lbl = k; }
            tlab[c] = (unsigned char)lbl;
        }

        // correctness check against prediction at (b, bn, cj, ci)
        {
            const float* p = pred + ((((size_t)b * NA_ + bn) * NH_ + cj) * NW_ + ci) * CH_;
            float px = p[1] + (float)ci;
            float py = p[2] + (float)cj;
            float pw = expf(p[4]) * c_AW[bn];
            float ph = expf(p[3]) * c_AH[bn];
            float gx1 = gx - gw * 0.5f, gx2 = gx + gw * 0.5f;
            float gy1 = gy - gh * 0.5f, gy2 = gy + gh * 0.5f;
            float px1 = px - pw * 0.5f, px2 = px + pw * 0.5f;
            float py1 = py - ph * 0.5f, py2 = py + ph * 0.5f;
            float iw = fmaxf(fminf(gx2, px2) - fmaxf(gx1, px1) + 1.0f, 0.0f);
            float ih = fmaxf(fminf(gy2, py2) - fmaxf(gy1, py1) + 1.0f, 0.0f);
            float inter = iw * ih;
            float ga = (gx2 - gx1 + 1.0f) * (gy2 - gy1 + 1.0f);
            float pa = (px2 - px1 + 1.0f) * (py2 - py1 + 1.0f);
            float iou = inter / (ga + pa - inter + 1e-16f);

            int amax = 0; float cm = p[5];
            for (int k = 1; k < NC_; ++k)
                if (p[5 + k] > cm) { cm = p[5 + k]; amax = k; }

            int lbl = 0; float lm = row[13];
            for (int k = 1; k < NC_; ++k)
                if (row[13 + k] > lm) { lm = row[13 + k]; lbl = k; }

            if ((iou > 0.5f) && (amax == lbl) && (p[0] > 0.5f)) nCor++;
        }
    }
    bGT[b]  = ts;
    bCor[b] = nCor;
}

// ---------------------------------------------------------------- main reduction
// sums: 0 nProp, 1 nCmf, 2 sumCmfBce, 3 sumMaskBce, 4 nM, 5 lx, 6 ly, 7 lw, 8 lh, 9 ce
// Grid is exactly NBLK2 x THREADS == NCELL (all lanes active -> async-to-LDS legal).
__global__ void reduce_cells_kernel(const float* __restrict__ pred,
                                    const unsigned char* __restrict__ flags,
                                    const unsigned char* __restrict__ tlab,
                                    const float* __restrict__ tx, const float* __restrict__ ty,
                                    const float* __restrict__ tw, const float* __restrict__ th,
                                    float* __restrict__ partials) {
    __shared__ float ldsConf[THREADS];
    const int tid  = threadIdx.x;
    const int cell = blockIdx.x * THREADS + tid;
    const float* p = pred + (size_t)cell * CH_;

    // ---- CDNA5 async global->LDS stage of the conf channel (ASYNCcnt path) ----
#if __has_builtin(__builtin_amdgcn_global_load_async_to_lds_b32)
    __builtin_amdgcn_global_load_async_to_lds_b32((gint_as*)p,
                                                  (lint_as*)&ldsConf[tid],
                                                  0, 0);
#else
    {
        unsigned lds_off = (unsigned)(size_t)(lint_as*)&ldsConf[tid];
        unsigned long long gaddr = (unsigned long long)(size_t)p;
        asm volatile("global_load_async_to_lds_b32 %0, %1, off"
                     :: "v"(lds_off), "v"(gaddr) : "memory");
    }
#endif

    // stream-ahead prefetch of the conf channel 8 tiles ahead (global_prefetch_b8)
    if (cell + 8 * THREADS < NCELL)
        __builtin_prefetch(pred + (size_t)(cell + 8 * THREADS) * CH_, 0, 0);

    // overlap: fetch per-cell flag while the async copy is in flight
    unsigned int f = flags[cell];

    // wait for our wave's async transfer
#if __has_builtin(__builtin_amdgcn_s_wait_asynccnt)
    __builtin_amdgcn_s_wait_asynccnt(0);
#else
    asm volatile("s_wait_asynccnt 0x0" ::: "memory");
#endif
    asm volatile("" ::: "memory");   // keep the LDS read below the wait
    float pc = ldsConf[tid];

    float v[NQ];
#pragma unroll
    for (int q = 0; q < NQ; ++q) v[q] = 0.0f;

    float m  = (float)(f & 1u);
    float cm = (float)((f >> 1) & 1u);

    v[0] = (pc > 0.0f) ? 1.0f : 0.0f;                               // nProposals
    float bce = fmaxf(pc, 0.0f) - pc * m + log1pf(expf(-fabsf(pc)));
    float cmf = (cm != m) ? 1.0f : 0.0f;
    v[1] = cmf;
    v[2] = cmf * bce;

    if (f & 1u) {                                                   // masked cell (rare)
        v[3] = bce;
        v[4] = 1.0f;
        float dx = p[1] - tx[cell]; v[5] = dx * dx;
        float dy = p[2] - ty[cell]; v[6] = dy * dy;
        float dw = p[4] - tw[cell]; v[7] = dw * dw;
        float dh = p[3] - th[cell]; v[8] = dh * dh;
        // stable log-softmax CE over the 40 classes
        float mx = p[5];
        for (int k = 1; k < NC_; ++k) mx = fmaxf(mx, p[5 + k]);
        float s = 0.0f;
        for (int k = 0; k < NC_; ++k) s += expf(p[5 + k] - mx);
        int lbl = (int)tlab[cell];
        v[9] = -((p[5 + lbl] - mx) - logf(s));
    }

    // single fixed-order LDS tree over all NQ quantities (deterministic)
    __shared__ float red[THREADS][NQ + 1];   // +1 pad: avoid bank conflicts
#pragma unroll
    for (int q = 0; q < NQ; ++q) red[tid][q] = v[q];
    __syncthreads();
    for (int s = THREADS / 2; s > 0; s >>= 1) {
        if (tid < s) {
#pragma unroll
            for (int q = 0; q < NQ; ++q) red[tid][q] += red[tid + s][q];
        }
        __syncthreads();
    }
    if (tid < NQ) partials[(size_t)blockIdx.x * NQ + tid] = red[0][tid];
}

// ---------------------------------------------------------------- finalize
__global__ void finalize_kernel(const float* __restrict__ partials,
                                const int* __restrict__ bGT, const int* __restrict__ bCor,
                                float* __restrict__ out) {
    const int tid = threadIdx.x;
    float acc[NQ];
#pragma unroll
    for (int q = 0; q < NQ; ++q) acc[q] = 0.0f;
    for (int i = tid; i < NBLK2; i += THREADS) {
#pragma unroll
        for (int q = 0; q < NQ; ++q) acc[q] += partials[(size_t)i * NQ + q];
    }
    __shared__ float red[THREADS][NQ + 1];
#pragma unroll
    for (int q = 0; q < NQ; ++q) red[tid][q] = acc[q];
    __syncthreads();
    for (int s = THREADS / 2; s > 0; s >>= 1) {
        if (tid < s) {
#pragma unroll
            for (int q = 0; q < NQ; ++q) red[tid][q] += red[tid + s][q];
        }
        __syncthreads();
    }
    if (tid == 0) {
        float S[NQ];
#pragma unroll
        for (int q = 0; q < NQ; ++q) S[q] = red[0][q];
        int nGT = 0, nCor = 0;
        for (int b = 0; b < NB_; ++b) { nGT += bGT[b]; nCor += bCor[b]; }
        float nM = S[4];
        float lx = S[5] / nM, ly = S[6] / nM, lw = S[7] / nM, lh = S[8] / nM;
        float lcoord = lx + ly + lw + lh;
        float lconf  = S[2] / S[1] + S[3] / nM;
        float lcls   = (1.0f / (float)NB_) * S[9] / nM;
        float loss   = lcoord + lconf + lcls;
        float nProp  = S[0];
        int   gtden  = nGT > 1 ? nGT : 1;
        float recall = (float)nCor / (float)gtden;
        float precision = (nProp > 0.0f) ? ((float)nCor / fmaxf(nProp, 1.0f)) : 1.0f;
        out[0] = loss;
        out[1] = lcoord;
        out[2] = lconf;
        out[3] = lcls;
        out[4] = recall;
        out[5] = precision;
    }
}

// ---------------------------------------------------------------- launch
extern "C" void kernel_launch(void* const* d_in, const int* in_sizes, int n_in,
                              void* d_out, int out_size, void* d_ws, size_t ws_size,
                              hipStream_t stream) {
    const float* pred   = (const float*)d_in[0];   // (16,5,96,96,45) f32
    const float* target = (const float*)d_in[1];   // (16,50,53) f32
    const int*   tsz    = (const int*)d_in[2];     // (16,) i32

    char* ws = (char*)d_ws;
    unsigned char* flags = (unsigned char*)ws;                 // NCELL bytes
    unsigned char* tlab  = (unsigned char*)(ws + NCELL);       // NCELL bytes
    float* tx = (float*)(ws + 2 * (size_t)NCELL);              // NCELL f32 each
    float* ty = tx + NCELL;
    float* tw = ty + NCELL;
    float* th = tw + NCELL;
    float* partials = th + NCELL;                              // NBLK2*NQ f32
    int* bGT  = (int*)(partials + (size_t)NBLK2 * NQ);         // 16 i32
    int* bCor = bGT + NB_;                                     // 16 i32

    init_flags_kernel<<<(NCELL / 4 + 255) / 256, 256, 0, stream>>>((unsigned int*)flags);
    build_targets_kernel<<<NB_, 1, 0, stream>>>(pred, target, tsz, flags, tlab,
                                                tx, ty, tw, th, bGT, bCor);
    reduce_cells_kernel<<<NBLK2, THREADS, 0, stream>>>(pred, flags, tlab,
                                                       tx, ty, tw, th, partials);
    finalize_kernel<<<1, THREADS, 0, stream>>>(partials, bGT, bCor, (float*)d_out);
}